// TinyLoRALayer_25838523252927
// MI455X (gfx1250) — compile-verified
//
#include <hip/hip_runtime.h>

// TinyLoRA fused kernel for MI455X (gfx1250, wave32).
//
// out = base + ((x @ B^T) * (P@v)) @ A^T      -- never materialize dW.
// ~1.1 GFLOP vs ~402 MB HBM traffic -> bandwidth bound (~17us @ 23.3 TB/s).
// Phase 1 (x @ B^T) uses V_WMMA_F32_16X16X4_F32: x staged through LDS
// (coalesced b128 global loads -> conflict-free ds_load_b64 fragments),
// B-fragments hoisted per chunk into registers with the P@v scale and the
// rank-8 -> N=16 zero pad folded into a single per-lane multiplier (no
// divergence, no global loads inside the WMMA loop).
// Phase 2 streams base/out with VALU FMAs (8 FMAs per output element).

typedef __attribute__((ext_vector_type(2))) float v2f;
typedef __attribute__((ext_vector_type(4))) float v4f;
typedef __attribute__((ext_vector_type(8))) float v8f;

#define D_DIM 4096
#define ROWS_TOTAL 8192   // B_SZ * SEQ = 2 * 4096
#define TILE_M 16
#define NWAVES 8
#define KPERWAVE 512      // 4096 / 8 waves
#define KCHUNK 64
#define NSTEP (KCHUNK / 4)
#define XS_STRIDE 68      // 64 + 4 pad: stride%64 == 4 -> conflict-free ds_load_b64

__global__ void __launch_bounds__(256)
lora_fused_kernel(const float* __restrict__ x,
                  const float* __restrict__ base,
                  const float* __restrict__ A,   // (4096, 8) row-major
                  const float* __restrict__ Bm,  // (8, 4096) row-major
                  const float* __restrict__ P,   // (8, 4)
                  const float* __restrict__ v,   // (4,)
                  float* __restrict__ out)
{
    __shared__ float xs[NWAVES][TILE_M][XS_STRIDE];  // 34816 B, per-wave private
    __shared__ float Tpart[NWAVES][16][16];          //  8192 B
    __shared__ float Tm[TILE_M][8];                  //   512 B

    const int tid  = threadIdx.x;
    const int w    = tid >> 5;    // wave id (wave32)
    const int lane = tid & 31;
    const long rowBase = (long)blockIdx.x * TILE_M;

    // Fragment coordinates (32-bit WMMA operand layouts, 16x16x4 f32):
    //   A-frag: lane L holds M = L%16, K = 2*(L/16) + {0,1}
    //   B-frag: lane L holds N = L%16, K = 2*(L/16) + {0,1}
    const int m    = lane & 15;          // M row (A-frag) / N col (B-frag)
    const int kofs = (lane >> 4) * 2;    // K sub-offset for upper half-wave
    const int kw0  = w * KPERWAVE;

    // Per-lane B-column scale: bs = (n < 8) ? (P@v)[n] : 0.
    // Folds both the LoRA scale and the rank-8 -> N=16 zero padding into one
    // multiplier; computed once, branch-free at use sites.
    float bs;
    {
        const int rq = m & 7;
        float s = 0.f;
        #pragma unroll
        for (int u = 0; u < 4; ++u) s += P[rq * 4 + u] * v[u];
        bs = (m < 8) ? s : 0.f;
    }
    const float* __restrict__ Brow = Bm + (long)(m & 7) * D_DIM;  // always valid

    // ---------------- Phase 1: T = (x_tile @ B^T) * scale  (WMMA f32) --------
    v8f acc = {};
    for (int kc = 0; kc < KPERWAVE / KCHUNK; ++kc) {
        const int kbase = kw0 + kc * KCHUNK;

        // Coalesced staging of 16 rows x 64 cols into this wave's LDS slab.
        #pragma unroll
        for (int i = 0; i < 8; ++i) {
            const int r  = i * 2 + (lane >> 4);
            const int c4 = lane & 15;
            v4f val = *(const v4f*)(x + (rowBase + r) * D_DIM + kbase + c4 * 4);
            *(v4f*)&xs[w][r][c4 * 4] = val;
        }

        // Hoist all 16 B-fragments for this chunk (unconditional, clause-able,
        // L2-hot; overlaps the ds_stores above). Scale+pad applied here.
        v2f bf[NSTEP];
        #pragma unroll
        for (int s = 0; s < NSTEP; ++s)
            bf[s] = *(const v2f*)(Brow + kbase + s * 4 + kofs);
        #pragma unroll
        for (int s = 0; s < NSTEP; ++s) { bf[s][0] *= bs; bf[s][1] *= bs; }

        // Hint next chunk toward near caches (global_prefetch_b8).
        if (kc < (KPERWAVE / KCHUNK) - 1) {
            __builtin_prefetch(x + (rowBase + (lane >> 4)) * D_DIM +
                               kbase + KCHUNK + (lane & 15) * 4, 0, 3);
        }

        // Pure ds_load + WMMA accumulate chain.
        #pragma unroll
        for (int s = 0; s < NSTEP; ++s) {
            v2f a = *(const v2f*)&xs[w][m][s * 4 + kofs];   // conflict-free b64
            acc = __builtin_amdgcn_wmma_f32_16x16x4_f32(
                false, a, false, bf[s], (short)0, acc, false, false);
        }
    }

    // Spill per-wave partial C to LDS.
    // C layout: lane L, vgpr g -> (M = g + 8*(L/16), N = L%16)
    {
        const int n   = lane & 15;
        const int mhi = (lane >> 4) * 8;
        #pragma unroll
        for (int g = 0; g < 8; ++g) Tpart[w][mhi + g][n] = acc[g];
    }
    __syncthreads();

    // Cross-wave reduce (scale already folded in): Tm (16 x 8)
    {
        const int M = tid >> 4;
        const int N = tid & 15;
        float s = 0.f;
        #pragma unroll
        for (int ww = 0; ww < NWAVES; ++ww) s += Tpart[ww][M][N];
        if (N < 8) Tm[M][N] = s;
    }
    __syncthreads();

    // ---------------- Phase 2: out = base + Tm @ A^T (streaming, BW-bound) ----
    // Each thread owns a 4-column strip; its 4x8 block of A loaded once per
    // strip (128B/lane, lane-contiguous); base/out accessed as coalesced
    // float4 per row; Tm rows are broadcast LDS reads.
    for (int sblk = 0; sblk < 4; ++sblk) {
        const int col = (tid + sblk * 256) * 4;   // 0..4092

        float areg[4][8];
        #pragma unroll
        for (int j = 0; j < 4; ++j) {
            v4f lo = *(const v4f*)(A + (long)(col + j) * 8);
            v4f hi = *(const v4f*)(A + (long)(col + j) * 8 + 4);
            #pragma unroll
            for (int r8 = 0; r8 < 4; ++r8) { areg[j][r8] = lo[r8]; areg[j][r8 + 4] = hi[r8]; }
        }

        #pragma unroll 4
        for (int r = 0; r < TILE_M; ++r) {
            v4f t0 = *(const v4f*)&Tm[r][0];
            v4f t1 = *(const v4f*)&Tm[r][4];
            float tr[8];
            #pragma unroll
            for (int r8 = 0; r8 < 4; ++r8) { tr[r8] = t0[r8]; tr[r8 + 4] = t1[r8]; }

            const long off = (rowBase + r) * D_DIM + col;
            v4f o = *(const v4f*)(base + off);
            #pragma unroll
            for (int j = 0; j < 4; ++j) {
                float s = 0.f;
                #pragma unroll
                for (int r8 = 0; r8 < 8; ++r8) s = fmaf(tr[r8], areg[j][r8], s);
                o[j] += s;
            }
            *(v4f*)(out + off) = o;
        }
    }
}

extern "C" void kernel_launch(void* const* d_in, const int* in_sizes, int n_in,
                              void* d_out, int out_size, void* d_ws, size_t ws_size,
                              hipStream_t stream) {
    (void)in_sizes; (void)n_in; (void)out_size; (void)d_ws; (void)ws_size;
    const float* x    = (const float*)d_in[0];
    const float* base = (const float*)d_in[1];
    const float* A    = (const float*)d_in[2];
    const float* B    = (const float*)d_in[3];
    const float* P    = (const float*)d_in[4];
    const float* v    = (const float*)d_in[5];
    float* out = (float*)d_out;

    dim3 grid(ROWS_TOTAL / TILE_M);   // 512 workgroups
    dim3 block(256);                  // 8 wave32 waves per WGP
    hipLaunchKernelGGL(lora_fused_kernel, grid, block, 0, stream,
                       x, base, A, B, P, v, out);
}